// WindowAttention_10917806867062
// MI455X (gfx1250) — compile-verified
//
#include <hip/hip_runtime.h>
#include <math.h>

typedef float v2f __attribute__((ext_vector_type(2)));
typedef float v8f __attribute__((ext_vector_type(8)));

#define WIN_N      64          // tokens per window (8x8)
#define NHEADS     8
#define HDIM       32
#define MODEL_DIM  256
#define QKV_STRIDE 768         // 3 * 256 floats per token
#define P_LD       66          // padded LDS row stride (8B aligned, bank-rotating)

static __device__ __forceinline__ float gelu_erf(float x) {
    return 0.5f * x * (1.0f + erff(x * 0.70710678118654752440f));
}

// ---------------------------------------------------------------------------
// Meta-MLP bias: [N,N,2] signed-log rel-pos -> Linear(2,256) -> erf-GELU ->
// Linear(256,8) -> bias[H][N][N]. Tiny (runs once, 128 KB result -> L2).
// ---------------------------------------------------------------------------
__global__ __launch_bounds__(256) void bias_mlp_kernel(
    const float* __restrict__ w1, const float* __restrict__ b1,
    const float* __restrict__ w2, const float* __restrict__ b2,
    float* __restrict__ biasOut)
{
    int idx = blockIdx.x * blockDim.x + threadIdx.x;   // 0..4095 = i*64+j
    if (idx >= WIN_N * WIN_N) return;
    int i = idx >> 6;
    int j = idx & 63;
    float d0 = (float)((i >> 3) - (j >> 3));           // row delta
    float d1 = (float)((i & 7)  - (j & 7));            // col delta
    float r0 = copysignf(log1pf(fabsf(d0)), d0);
    float r1 = copysignf(log1pf(fabsf(d1)), d1);

    float acc[NHEADS];
#pragma unroll
    for (int h = 0; h < NHEADS; ++h) acc[h] = b2[h];

    for (int k = 0; k < 256; ++k) {
        float x = fmaf(r0, w1[k], fmaf(r1, w1[256 + k], b1[k]));
        float g = gelu_erf(x);
#pragma unroll
        for (int h = 0; h < NHEADS; ++h)
            acc[h] = fmaf(g, w2[k * NHEADS + h], acc[h]);
    }
#pragma unroll
    for (int h = 0; h < NHEADS; ++h)
        biasOut[h * (WIN_N * WIN_N) + idx] = acc[h];
}

// ---------------------------------------------------------------------------
// Window attention: grid (B, 4), block 64 (= 2 wave32s, one head per wave).
// fp32 WMMA 16x16x4 for both GEMMs; logits staged in LDS for softmax and
// the C-layout -> A-layout fragment reshuffle; normalization deferred to
// after the PV GEMM using row sums parked in the LDS pad column.
// ---------------------------------------------------------------------------
__global__ __launch_bounds__(64) void win_attn_kernel(
    const float* __restrict__ qkv, const float* __restrict__ bias,
    float* __restrict__ out)
{
    __shared__ float sP[2][WIN_N * P_LD];              // 33.8 KB

    const int wave = threadIdx.x >> 5;                 // 0..1
    const int lane = threadIdx.x & 31;
    const int hsel = lane >> 4;                        // K-pair / M-half select
    const int lm   = lane & 15;
    const int b    = blockIdx.x;
    const int head = blockIdx.y * 2 + wave;            // 0..7

    float* P = sP[wave];

    const float* Qb = qkv + (size_t)b * (WIN_N * QKV_STRIDE) + head * HDIM;
    const float* Kb = Qb + 256;
    const float* Vb = Qb + 512;
    const float* Bh = bias + head * (WIN_N * WIN_N);

    __builtin_prefetch(Qb, 0, 1);                      // global_prefetch_b8

    // ---------------- S = Q K^T  (64x64, K=32) ----------------
    // f32 16x16x4 A/B fragment: lane = idx%16 + 16*(k>>1), vgpr = k&1
    // -> one aligned b64 load per fragment straight from the qkv tensor.
    v8f acc[4][4] = {};
    for (int k4 = 0; k4 < 8; ++k4) {
        const int kc = k4 * 4 + hsel * 2;
        v2f af[4], bf[4];
#pragma unroll
        for (int t = 0; t < 4; ++t) {
            af[t] = *(const v2f*)(Qb + (t * 16 + lm) * QKV_STRIDE + kc);
            bf[t] = *(const v2f*)(Kb + (t * 16 + lm) * QKV_STRIDE + kc);
        }
#pragma unroll
        for (int mt = 0; mt < 4; ++mt)
#pragma unroll
            for (int nt = 0; nt < 4; ++nt)
                acc[mt][nt] = __builtin_amdgcn_wmma_f32_16x16x4_f32(
                    false, af[mt], false, bf[nt], (short)0, acc[mt][nt],
                    false, false);
    }

    // scale + MLP bias, spill logits to LDS (C layout: m = 16*mt+8*hsel+g)
    constexpr float SCALE = 0.17677669529663687f;      // 32^-0.5
#pragma unroll
    for (int mt = 0; mt < 4; ++mt)
#pragma unroll
        for (int nt = 0; nt < 4; ++nt)
#pragma unroll
            for (int g = 0; g < 8; ++g) {
                int m = mt * 16 + hsel * 8 + g;
                int n = nt * 16 + lm;
                P[m * P_LD + n] = acc[mt][nt][g] * SCALE + Bh[m * WIN_N + n];
            }
    __syncthreads();

    // ---------------- softmax (unnormalized) ----------------
    // 2 rows per lane; row stride 66 -> 2-bank rotation, conflict-free.
#pragma unroll
    for (int rr = 0; rr < 2; ++rr) {
        int r = lane + rr * 32;
        float* row = P + r * P_LD;
        float mx = -3.402823466e+38f;
        for (int jj = 0; jj < WIN_N; ++jj) mx = fmaxf(mx, row[jj]);
        float s = 0.0f;
        for (int jj = 0; jj < WIN_N; ++jj) {
            float e = __expf(row[jj] - mx);
            row[jj] = e;
            s += e;
        }
        row[64] = s;                                   // row sum in pad column
    }
    __syncthreads();

    // ---------------- O = P V  (64x32, K=64) ----------------
    v8f o[4][2] = {};
    for (int k4 = 0; k4 < 16; ++k4) {
        const int kr = k4 * 4 + hsel * 2;
        v2f af[4];
#pragma unroll
        for (int t = 0; t < 4; ++t)                    // ds_load_b64 (aligned)
            af[t] = *(const v2f*)&P[(t * 16 + lm) * P_LD + kr];
        v2f bf[2];
#pragma unroll
        for (int nt = 0; nt < 2; ++nt) {               // B[k][n] = V[k][n]
            v2f t;
            t.x = Vb[(kr + 0) * QKV_STRIDE + nt * 16 + lm];
            t.y = Vb[(kr + 1) * QKV_STRIDE + nt * 16 + lm];
            bf[nt] = t;
        }
#pragma unroll
        for (int mt = 0; mt < 4; ++mt)
#pragma unroll
            for (int nt = 0; nt < 2; ++nt)
                o[mt][nt] = __builtin_amdgcn_wmma_f32_16x16x4_f32(
                    false, af[mt], false, bf[nt], (short)0, o[mt][nt],
                    false, false);
    }

    // ---------------- normalize + store ----------------
    float* Ob = out + (size_t)b * (WIN_N * MODEL_DIM) + head * HDIM;
#pragma unroll
    for (int mt = 0; mt < 4; ++mt)
#pragma unroll
        for (int g = 0; g < 8; ++g) {
            int m = mt * 16 + hsel * 8 + g;
            float inv = 1.0f / P[m * P_LD + 64];
#pragma unroll
            for (int nt = 0; nt < 2; ++nt) {
                int d = nt * 16 + lm;
                Ob[m * MODEL_DIM + d] = o[mt][nt][g] * inv;
            }
        }
}

// ---------------------------------------------------------------------------
extern "C" void kernel_launch(void* const* d_in, const int* in_sizes, int n_in,
                              void* d_out, int out_size, void* d_ws, size_t ws_size,
                              hipStream_t stream) {
    const float* qkv = (const float*)d_in[0];
    const float* w1  = (const float*)d_in[1];
    const float* b1  = (const float*)d_in[2];
    const float* w2  = (const float*)d_in[3];
    const float* b2  = (const float*)d_in[4];
    float* out  = (float*)d_out;
    float* bias = (float*)d_ws;                        // 8*64*64 floats = 128 KB

    const int B = in_sizes[0] / (WIN_N * QKV_STRIDE);  // 2048

    bias_mlp_kernel<<<dim3((WIN_N * WIN_N + 255) / 256), dim3(256), 0, stream>>>(
        w1, b1, w2, b2, bias);
    win_attn_kernel<<<dim3(B, NHEADS / 2), dim3(64), 0, stream>>>(
        qkv, bias, out);
}